// ScatterAvgBlock_41420664602706
// MI455X (gfx1250) — compile-verified
//
#include <hip/hip_runtime.h>
#include <hip/hip_bf16.h>
#include <stdint.h>

// Problem constants (fixed by the reference's setup_inputs)
#define N_BLK 1024
#define C_CH  128
#define BH    16
#define BW    16
#define H_IMG 512
#define W_IMG 512
#define HW    (H_IMG * W_IMG)      // 262144
#define HW4   (HW / 4)             // 65536 = 2^16

typedef float v4f __attribute__((ext_vector_type(4)));

// ---------------------------------------------------------------------------
// Kernel 1: zero the per-pixel overlap-count buffer (float[HW] in d_ws).
// ---------------------------------------------------------------------------
__global__ void zero_cnt_kernel(float4* __restrict__ cnt4) {
    int t = blockIdx.x * blockDim.x + threadIdx.x;   // 65536 threads
    cnt4[t] = make_float4(0.f, 0.f, 0.f, 0.f);
}

// ---------------------------------------------------------------------------
// Kernel 2: scatter-add 1.0 per covered pixel. One thread per (block, pixel).
// 262144 f32 atomics into a 1 MiB buffer -> entirely L2-resident, negligible.
// ---------------------------------------------------------------------------
__global__ void count_scatter_kernel(const int* __restrict__ aidx,
                                     float* __restrict__ cnt) {
    int t   = blockIdx.x * blockDim.x + threadIdx.x; // N_BLK*256 threads
    int n   = t >> 8;
    int pix = t & 255;
    int sy  = aidx[2 * n];
    int sx  = aidx[2 * n + 1];
    int r   = pix >> 4;
    int cc  = pix & 15;
    atomicAdd(&cnt[(sy + r) * W_IMG + (sx + cc)], 1.0f);
}

// ---------------------------------------------------------------------------
// Kernel 3: out = (cnt > 0) ? 0 : base.   Vectorized float4.  `base` is read
// exactly once -> non-temporal load (don't evict cnt/out from L2); `out` is
// written RT so it stays L2-resident for the atomic scatter pass.
// ---------------------------------------------------------------------------
__global__ void init_out_kernel(const v4f* __restrict__ base4,
                                const float4* __restrict__ cnt4,
                                float4* __restrict__ out4) {
    int t  = blockIdx.x * blockDim.x + threadIdx.x;  // C_CH*HW4 threads
    int p4 = t & (HW4 - 1);
    float4 cn = cnt4[p4];
    v4f    b  = __builtin_nontemporal_load(base4 + t);   // NT: streamed once
    float4 o;
    o.x = (cn.x > 0.f) ? 0.f : b.x;
    o.y = (cn.y > 0.f) ? 0.f : b.y;
    o.z = (cn.z > 0.f) ? 0.f : b.z;
    o.w = (cn.w > 0.f) ? 0.f : b.w;
    out4[t] = o;
}

// ---------------------------------------------------------------------------
// Kernel 4: the heavy scatter.  One wave32 per (block n, channel c):
//   - async-DMA the 1 KiB x-tile into LDS (2x global_load_async_to_lds_b128,
//     each lane moves 16 B; INST_OFFSET applies to BOTH the global and LDS
//     address, so one vaddr/ldsaddr pair covers both halves).  th:TH_LOAD_NT
//     because x is streamed exactly once -- keeps the 128 MiB output image
//     resident in the 192 MB L2 so the atomics below stay L2-hits,
//   - s_wait_asynccnt 0,
//   - read back 2x float4 per lane from LDS (ds_load_b128),
//   - 8 global_atomic_add_f32 per lane into the L2-resident output image.
// Workgroup = 8 waves = 8 channels of one block; 16 workgroups per block.
// LDS: 8 KiB/workgroup.
// ---------------------------------------------------------------------------
__global__ void scatter_add_kernel(const float* __restrict__ x,
                                   const int* __restrict__ aidx,
                                   float* __restrict__ out) {
    __shared__ __align__(16) float tile[8 * 256];   // 1 KiB per wave

    const int wave = threadIdx.x >> 5;
    const int lane = threadIdx.x & 31;
    const int wg   = blockIdx.x;                    // 0 .. N_BLK*16-1
    const int n    = wg >> 4;
    const int c    = ((wg & 15) << 3) | wave;       // 0..127

    const int sy = aidx[2 * n];
    const int sx = aidx[2 * n + 1];

    const float* gsrc = x + ((size_t)(n * C_CH + c) << 8);  // 256 floats

    // LDS byte offset for this lane's 16 B slot (ptrtoint escapes `tile`,
    // and the asm "memory" clobber orders the later LDS reads after the DMA).
    unsigned ldsOff = (unsigned)(uintptr_t)tile
                    + ((unsigned)wave << 10) + ((unsigned)lane << 4);
    unsigned vOff   = (unsigned)lane << 4;

    asm volatile(
        "global_load_async_to_lds_b128 %0, %1, %2 offset:0 th:TH_LOAD_NT\n\t"
        "global_load_async_to_lds_b128 %0, %1, %2 offset:512 th:TH_LOAD_NT"
        :: "v"(ldsOff), "v"(vOff), "s"(gsrc)
        : "memory");
    asm volatile("s_wait_asynccnt 0" ::: "memory");

    float* dst = out + (size_t)c * HW + (size_t)sy * W_IMG + sx;
    const float4* tq = (const float4*)(tile + (wave << 8));

#pragma unroll
    for (int half = 0; half < 2; ++half) {
        int    e = (half << 7) + (lane << 2);       // element 0..255
        float4 v = tq[(half << 5) + lane];          // ds_load_b128
        int    r  = e >> 4;
        int    cc = e & 15;
        float* p  = dst + r * W_IMG + cc;           // 4 contiguous pixels
        atomicAdd(p + 0, v.x);
        atomicAdd(p + 1, v.y);
        atomicAdd(p + 2, v.z);
        atomicAdd(p + 3, v.w);
    }
}

// ---------------------------------------------------------------------------
// Kernel 5: normalize covered pixels: out /= cnt (cnt>0).  Quads that are
// fully uncovered skip the out load AND store (saves ~37% of the traffic).
// out is hot in L2 after the atomic pass, so these are mostly L2 hits.
// ---------------------------------------------------------------------------
__global__ void normalize_kernel(const float4* __restrict__ cnt4,
                                 float4* __restrict__ out4) {
    int t  = blockIdx.x * blockDim.x + threadIdx.x;  // C_CH*HW4 threads
    int p4 = t & (HW4 - 1);
    float4 cn = cnt4[p4];
    if (cn.x > 0.f || cn.y > 0.f || cn.z > 0.f || cn.w > 0.f) {
        float4 v = out4[t];
        if (cn.x > 0.f) v.x /= cn.x;
        if (cn.y > 0.f) v.y /= cn.y;
        if (cn.z > 0.f) v.z /= cn.z;
        if (cn.w > 0.f) v.w /= cn.w;
        out4[t] = v;
    }
}

// ---------------------------------------------------------------------------
extern "C" void kernel_launch(void* const* d_in, const int* in_sizes, int n_in,
                              void* d_out, int out_size, void* d_ws, size_t ws_size,
                              hipStream_t stream) {
    const float* x    = (const float*)d_in[0];  // [N, C, 16, 16]
    const float* base = (const float*)d_in[1];  // [1, C, 512, 512]
    const int*   aidx = (const int*)d_in[2];    // [N, 2]
    float*       out  = (float*)d_out;          // [1, C, 512, 512]
    float*       cnt  = (float*)d_ws;           // HW floats = 1 MiB scratch

    (void)in_sizes; (void)n_in; (void)out_size; (void)ws_size;

    // 1) cnt = 0
    zero_cnt_kernel<<<HW4 / 256, 256, 0, stream>>>((float4*)cnt);
    // 2) cnt += 1 per covered pixel
    count_scatter_kernel<<<(N_BLK * 256) / 256, 256, 0, stream>>>(aidx, cnt);
    // 3) out = covered ? 0 : base
    init_out_kernel<<<(C_CH * HW4) / 256, 256, 0, stream>>>(
        (const v4f*)base, (const float4*)cnt, (float4*)out);
    // 4) out += x (scattered, atomic, via async-LDS staging, NT reads of x)
    scatter_add_kernel<<<N_BLK * 16, 256, 0, stream>>>(x, aidx, out);
    // 5) out /= cnt where covered
    normalize_kernel<<<(C_CH * HW4) / 256, 256, 0, stream>>>(
        (const float4*)cnt, (float4*)out);
}